// alignLearningLoss_48558900248644
// MI455X (gfx1250) — compile-verified
//
#include <hip/hip_runtime.h>

// Problem constants (fixed by the reference).
#define B_N   4096
#define M_N   3
#define D_N   256
#define ROWST (M_N * D_N)      // 768 floats between consecutive samples
#define TILE_ROWS 32
#define AST   260              // padded LDS row stride (floats): conflict-free + 16B aligned
#define NEG_INF (-1e30f)

typedef __attribute__((ext_vector_type(2))) float v2f;
typedef __attribute__((ext_vector_type(4))) float v4f;
typedef __attribute__((ext_vector_type(8))) float v8f;

__global__ void init_ws_kernel(int* hist) {
    if (threadIdx.x < 32) hist[threadIdx.x] = 0;
}

__global__ void hist_kernel(const int* __restrict__ labels, int* __restrict__ hist) {
    int i = blockIdx.x * blockDim.x + threadIdx.x;
    if (i < B_N) atomicAdd(&hist[labels[i]], 1);
}

// Grid: (B_N/32, M_N). Block: 256 threads = 8 waves (wave32).
__global__ __launch_bounds__(256) void simloss_main(
    const float* __restrict__ tokens, const int* __restrict__ labels,
    const int* __restrict__ hist, float* __restrict__ perbuf) {

    __shared__ float Alds[TILE_ROWS * AST];          // 33280 B
    __shared__ float part_mx[8][TILE_ROWS];
    __shared__ float part_su[8][TILE_ROWS];
    __shared__ float part_ps[8][TILE_ROWS];

    const int rowbase = blockIdx.x * TILE_ROWS;
    const int m       = blockIdx.y;
    const int tid     = threadIdx.x;

    // Stage the 32-row A tile (modality m) into LDS as float4s.
    for (int c = 0; c < 8; ++c) {
        int f   = c * 256 + tid;          // float4 index, 0..2047
        int row = f >> 6;                 // 64 float4 per row
        int d4  = f & 63;
        v4f val = *(const v4f*)(tokens + (size_t)(rowbase + row) * ROWST + m * D_N + d4 * 4);
        *(v4f*)(&Alds[row * AST + d4 * 4]) = val;
    }
    __syncthreads();

    const int lane  = tid & 31;
    const int wave  = tid >> 5;
    const int n     = lane & 15;          // column within 16-wide tile
    const int khalf = lane >> 4;          // K-half select for A/B fragments
    const int kh2   = khalf * 2;

    // Per-lane running stats: 2 row sub-tiles x 8 C-register rows.
    float mx[2][8], su[2][8], ps[2][8];
    int   labrow[2][8];
    for (int s = 0; s < 2; ++s)
        for (int v = 0; v < 8; ++v) {
            mx[s][v] = NEG_INF; su[s][v] = 0.f; ps[s][v] = 0.f;
            labrow[s][v] = labels[rowbase + s * 16 + v + 8 * khalf];
        }

    // A-fragment pattern: lane L -> row (L&15), floats [kb + 2*(L>>4), +1]
    const float* Arow0 = &Alds[n * AST + kh2];
    const float* Arow1 = &Alds[(16 + n) * AST + kh2];

    for (int ct = wave; ct < B_N / 16; ct += 8) {
        const int colbase = ct * 16;
        const int gcol    = colbase + n;
        const int labcol  = labels[gcol];
        // B-fragment (4x16 K-major): lane L -> col (L&15), floats [kb + 2*(L>>4), +1]
        const float* Bp = tokens + (size_t)gcol * ROWST + m * D_N + kh2;

        v8f acc[2] = {};
        #pragma unroll 8
        for (int ks = 0; ks < 64; ++ks) {
            const int kb = ks * 4;
            v2f a0 = *(const v2f*)(Arow0 + kb);
            v2f a1 = *(const v2f*)(Arow1 + kb);
            v2f bf = *(const v2f*)(Bp + kb);
            acc[0] = __builtin_amdgcn_wmma_f32_16x16x4_f32(
                false, a0, false, bf, (short)0, acc[0], false, false);
            acc[1] = __builtin_amdgcn_wmma_f32_16x16x4_f32(
                false, a1, false, bf, (short)0, acc[1], false, false);
        }

        // Fold the 16x16 products into online softmax stats + positive sums.
        #pragma unroll
        for (int s = 0; s < 2; ++s) {
            #pragma unroll
            for (int v = 0; v < 8; ++v) {
                float val  = acc[s][v];
                int   grow = rowbase + s * 16 + v + 8 * khalf;
                bool  self = (grow == gcol);
                float vm   = self ? NEG_INF : val;
                float om   = mx[s][v];
                float nm   = fmaxf(om, vm);
                su[s][v]   = su[s][v] * __expf(om - nm) + __expf(vm - nm);
                mx[s][v]   = nm;
                if (!self && (labrow[s][v] == labcol)) ps[s][v] += val;
            }
        }
    }

    // Reduce over the 16 columns held by each 16-lane half (rows differ per half).
    #pragma unroll
    for (int s = 0; s < 2; ++s) {
        #pragma unroll
        for (int v = 0; v < 8; ++v) {
            float M = mx[s][v], S = su[s][v], P = ps[s][v];
            for (int off = 8; off >= 1; off >>= 1) {
                float oM = __shfl_xor(M, off, 16);
                float oS = __shfl_xor(S, off, 16);
                float oP = __shfl_xor(P, off, 16);
                float nm = fmaxf(M, oM);
                S = S * __expf(M - nm) + oS * __expf(oM - nm);
                M = nm;
                P += oP;
            }
            if (n == 0) {
                int r = s * 16 + v + 8 * khalf;
                part_mx[wave][r] = M;
                part_su[wave][r] = S;
                part_ps[wave][r] = P;
            }
        }
    }
    __syncthreads();

    // Combine the 8 waves' partials; emit per[i,m].
    if (tid < TILE_ROWS) {
        float M = NEG_INF, S = 0.f, P = 0.f;
        for (int w = 0; w < 8; ++w) {
            float oM = part_mx[w][tid], oS = part_su[w][tid], oP = part_ps[w][tid];
            float nm = fmaxf(M, oM);
            S = S * __expf(M - nm) + oS * __expf(oM - nm);
            M = nm;
            P += oP;
        }
        float logZ = logf(S) + M;
        int   grow = rowbase + tid;
        int   cnt  = hist[labels[grow]] - 1;
        float per  = (cnt > 0) ? (P - (float)cnt * logZ) / (float)cnt : 0.f;
        perbuf[m * B_N + grow] = per;
    }
}

__global__ __launch_bounds__(256) void finalize_kernel(
    const float* __restrict__ perbuf, const int* __restrict__ labels,
    const int* __restrict__ hist, float* __restrict__ out) {
    __shared__ float sred[256];
    __shared__ int   ired[256];
    int tid = threadIdx.x;
    float a = 0.f;
    for (int i = tid; i < B_N * M_N; i += 256) a += perbuf[i];
    int vc = 0;
    for (int i = tid; i < B_N; i += 256) vc += (hist[labels[i]] > 1) ? 1 : 0;
    sred[tid] = a; ired[tid] = vc;
    __syncthreads();
    for (int s = 128; s > 0; s >>= 1) {
        if (tid < s) { sred[tid] += sred[tid + s]; ired[tid] += ired[tid + s]; }
        __syncthreads();
    }
    if (tid == 0) out[0] = -sred[0] / (float)(ired[0] * M_N);
}

extern "C" void kernel_launch(void* const* d_in, const int* in_sizes, int n_in,
                              void* d_out, int out_size, void* d_ws, size_t ws_size,
                              hipStream_t stream) {
    const float* tokens = (const float*)d_in[0];
    const int*   labels = (const int*)d_in[1];
    int*   hist   = (int*)d_ws;
    float* perbuf = (float*)d_ws + 32;   // 12288 floats of scratch
    float* out    = (float*)d_out;

    init_ws_kernel<<<1, 32, 0, stream>>>(hist);
    hist_kernel<<<B_N / 256, 256, 0, stream>>>(labels, hist);
    simloss_main<<<dim3(B_N / TILE_ROWS, M_N), 256, 0, stream>>>(tokens, labels, hist, perbuf);
    finalize_kernel<<<1, 256, 0, stream>>>(perbuf, labels, hist, out);
}